// ViewMorphing_50826642981119
// MI455X (gfx1250) — compile-verified
//
#include <hip/hip_runtime.h>
#include <math.h>

#define IMG 224
#define HW  (IMG * IMG)
#define TH  32              // h-extent of tile == lanes of a wave (gather coalescing)
#define TW  16              // w-extent of tile
#define PAD (TW + 1)        // LDS row stride in dwords for TDM-loaded tiles
#define PAD_INTERVAL_ENC 3  // pad every (1 << (enc+1)) = 16 dwords
#define PAD_AMOUNT_ENC   0  // pad amount = 1 dword

typedef unsigned int       u32;
typedef unsigned long long u64;
typedef __attribute__((ext_vector_type(4))) u32 v4u;
typedef __attribute__((ext_vector_type(8))) int v8i;
typedef __attribute__((ext_vector_type(4))) int v4i;

// Build a TDM descriptor for a (TW x TH x tile_d2) tile of 4-byte elements in a
// tensor with row pitch IMG and plane pitch dim1_stride, and issue either
// tensor_load_to_lds (with LDS padding for conflict-free transposed reads) or
// tensor_store_from_lds (contiguous LDS, no padding).
template <bool IS_STORE>
__device__ __forceinline__ void tdm_tile(u32 lds_off, u64 gaddr,
                                         u32 tile_d2, u32 tensor_d2,
                                         u32 dim1_stride) {
  v4u g0;
  g0.x = 1u;                                                  // count=1 (valid user D#)
  g0.y = lds_off;                                             // lds_addr (bytes)
  g0.z = (u32)gaddr;                                          // global_addr[31:0]
  g0.w = (u32)((gaddr >> 32) & 0x1FFFFFFull) | (2u << 30);    // global_addr[56:32] | type=2

  u32 d0 = (2u << 16);                                        // data_size = 4 bytes
  if (!IS_STORE) {                                            // LDS padding: loads only
    d0 |= (1u << 20) | ((u32)PAD_INTERVAL_ENC << 22) | ((u32)PAD_AMOUNT_ENC << 25);
  }
  v8i g1;
  g1[0] = (int)d0;
  g1[1] = (int)(((u32)IMG & 0xFFFFu) << 16);     // barrier_addr=0 | tensor_dim0 lo16
  g1[2] = (int)(((u32)IMG & 0xFFFFu) << 16);     // tensor_dim0 hi16=0 | tensor_dim1 lo16
  g1[3] = (int)((u32)TW << 16);                  // tensor_dim1 hi16=0 | tile_dim0
  g1[4] = (int)((u32)TH | (tile_d2 << 16));      // tile_dim1 | tile_dim2
  g1[5] = (int)IMG;                              // tensor_dim0_stride lo32
  g1[6] = (int)((dim1_stride & 0xFFFFu) << 16);  // stride0 hi16=0 | tensor_dim1_stride lo16
  g1[7] = (int)(dim1_stride >> 16);              // tensor_dim1_stride [47:16]

  v4i g2;
  g2[0] = (int)tensor_d2;                        // tensor_dim2 (tiles are interior)
  g2[1] = 0; g2[2] = 0; g2[3] = 0;
  v4i g3; g3[0] = 0; g3[1] = 0; g3[2] = 0; g3[3] = 0;

#if defined(__clang_major__) && (__clang_major__ >= 23)
  v8i g4; g4[0]=0; g4[1]=0; g4[2]=0; g4[3]=0; g4[4]=0; g4[5]=0; g4[6]=0; g4[7]=0;
  if (IS_STORE) __builtin_amdgcn_tensor_store_from_lds(g0, g1, g2, g3, g4, 0);
  else          __builtin_amdgcn_tensor_load_to_lds (g0, g1, g2, g3, g4, 0);
#else
  if (IS_STORE) __builtin_amdgcn_tensor_store_from_lds(g0, g1, g2, g3, 0);
  else          __builtin_amdgcn_tensor_load_to_lds (g0, g1, g2, g3, 0);
#endif
}

// Bilinear gather over 3 channels, replicating the reference exactly:
// 4 corners (floor/ceil both axes, independently clamped flat index nx + 224*ny),
// weights (1-|px-nx|)*(1-|py-ny|) summed over all 4 corners.
__device__ __forceinline__ void bilerp3(const float* __restrict__ im,
                                        float px, float py, float r[3]) {
  float fx = floorf(px), cx = ceilf(px);
  float fy = floorf(py), cy = ceilf(py);
  float wfx = 1.0f - fabsf(px - fx), wcx = 1.0f - fabsf(px - cx);
  float wfy = 1.0f - fabsf(py - fy), wcy = 1.0f - fabsf(py - cy);

  int iff = (int)(fx + 224.0f * fy);
  int icf = (int)(cx + 224.0f * fy);
  int ifc = (int)(fx + 224.0f * cy);
  int icc = (int)(cx + 224.0f * cy);
  iff = min(max(iff, 0), HW - 1);
  icf = min(max(icf, 0), HW - 1);
  ifc = min(max(ifc, 0), HW - 1);
  icc = min(max(icc, 0), HW - 1);

  float wff = wfx * wfy, wcf = wcx * wfy, wfc = wfx * wcy, wcc = wcx * wcy;
#pragma unroll
  for (int c = 0; c < 3; ++c) {
    const float* p = im + (u64)c * HW;
    r[c] = wff * p[iff] + wcf * p[icf] + wfc * p[ifc] + wcc * p[icc];
  }
}

__global__ __launch_bounds__(TH * TW) void ViewMorphing_kernel(
    const float* __restrict__ im1, const float* __restrict__ im2,
    const float* __restrict__ C,   const float* __restrict__ M1,
    const float* __restrict__ M2,  float* __restrict__ out) {
  __shared__ float sC  [2 * TH * PAD];   // TDM-padded input tiles
  __shared__ float sM1 [TH * PAD];
  __shared__ float sM2 [TH * PAD];
  __shared__ float sOut[3 * TH * TW];    // contiguous [c][h][w] for TDM store

  const int tx = threadIdx.x;            // h within tile (= lane: coalesced gathers)
  const int ty = threadIdx.y;            // w within tile
  const int w0 = blockIdx.x * TW;
  const int h0 = blockIdx.y * TH;
  const int n  = blockIdx.z;

  if (ty == 0) {  // exactly wave 0 issues the async input tile DMAs
    u64 baseC  = (u64)(uintptr_t)C  + 4ull * ((u64)n * 2 * HW + (u64)h0 * IMG + (u64)w0);
    u64 baseM1 = (u64)(uintptr_t)M1 + 4ull * ((u64)n * HW     + (u64)h0 * IMG + (u64)w0);
    u64 baseM2 = (u64)(uintptr_t)M2 + 4ull * ((u64)n * HW     + (u64)h0 * IMG + (u64)w0);
    tdm_tile<false>((u32)(uintptr_t)sC,  baseC,  2u, 2u, (u32)HW);  // 3D: 2 flow channels
    tdm_tile<false>((u32)(uintptr_t)sM1, baseM1, 0u, 0u, 0u);       // 2D mask tile
    tdm_tile<false>((u32)(uintptr_t)sM2, baseM2, 0u, 0u, 0u);       // 2D mask tile
    __builtin_amdgcn_s_wait_tensorcnt(0);
  }
  __syncthreads();

  const int h = h0 + tx;
  const int w = w0 + ty;

  // Transposed LDS reads: lane stride = PAD = 17 dwords -> conflict-free.
  float c0 = sC [( 0 * TH + tx) * PAD + ty];
  float c1 = sC [( 1 * TH + tx) * PAD + ty];
  float m1 = sM1[tx * PAD + ty];
  float m2 = sM2[tx * PAD + ty];

  const float* i1 = im1 + (u64)n * 3 * HW;
  const float* i2 = im2 + (u64)n * 3 * HW;

  float a[3], b[3];
  bilerp3(i1, (float)h + c0, (float)w + c1, a);   // sample im1 at q + C
  bilerp3(i2, (float)h - c0, (float)w - c1, b);   // sample im2 at q - C

#pragma unroll
  for (int c = 0; c < 3; ++c)
    sOut[(c * TH + tx) * TW + ty] = a[c] * m1 + b[c] * m2;
  __syncthreads();

  if (ty == 0) {  // wave 0 DMAs the finished tile out; S_ENDPGM waits TENSORcnt
    u64 baseO = (u64)(uintptr_t)out + 4ull * ((u64)n * 3 * HW + (u64)h0 * IMG + (u64)w0);
    tdm_tile<true>((u32)(uintptr_t)sOut, baseO, 3u, 3u, (u32)HW);
  }
}

extern "C" void kernel_launch(void* const* d_in, const int* in_sizes, int n_in,
                              void* d_out, int out_size, void* d_ws, size_t ws_size,
                              hipStream_t stream) {
  (void)n_in; (void)out_size; (void)d_ws; (void)ws_size;
  const float* im1 = (const float*)d_in[0];
  const float* im2 = (const float*)d_in[1];
  const float* C   = (const float*)d_in[2];
  const float* M1  = (const float*)d_in[3];
  const float* M2  = (const float*)d_in[4];
  float* out = (float*)d_out;

  const int N = in_sizes[0] / (3 * HW);          // 64
  dim3 block(TH, TW);                            // (32,16): lanes run along h
  dim3 grid(IMG / TW, IMG / TH, N);              // (14, 7, 64)
  ViewMorphing_kernel<<<grid, block, 0, stream>>>(im1, im2, C, M1, M2, out);
}